// GaussianUpsamplingModule_24017457119363
// MI455X (gfx1250) — compile-verified
//
#include <hip/hip_runtime.h>
#include <math.h>

// Problem constants (match reference)
#define Bn 32
#define Ln 512
#define Dn 512
#define LOG2PI 1.8378770664093453f   // log(2*pi)

typedef __attribute__((ext_vector_type(2))) float v2f;
typedef __attribute__((ext_vector_type(8))) float v8f;

// ---------------------------------------------------------------------------
// K1: conv1d(dur/en/pitch) + x_new = x + en_e + pi_e ; ranges = softplus(lin)
// one block (256 thr) per (b,l); 2 d-elements per thread; LDS tree-reduce dot
// ---------------------------------------------------------------------------
__global__ __launch_bounds__(256)
void k1_features(const float* __restrict__ x,
                 const float* __restrict__ df,
                 const float* __restrict__ en,
                 const float* __restrict__ pi,
                 const int*   __restrict__ lens,
                 const float* __restrict__ wd, const float* __restrict__ bd,
                 const float* __restrict__ we, const float* __restrict__ be,
                 const float* __restrict__ wp, const float* __restrict__ bp,
                 const float* __restrict__ wlin, const float* __restrict__ blin,
                 float* __restrict__ xnew,
                 float* __restrict__ inv_r,
                 float* __restrict__ log_r)
{
    const int bl = blockIdx.x;
    const int b  = bl / Ln;
    const int l  = bl % Ln;
    __shared__ float red[256];

    // neighbor scalars (zero padding at the ends), cross-correlation k=0 -> l-1
    const float dfm = (l > 0)      ? df[b*Ln + l - 1] : 0.f;
    const float df0 =                df[b*Ln + l];
    const float dfp = (l < Ln - 1) ? df[b*Ln + l + 1] : 0.f;
    const float enm = (l > 0)      ? en[b*Ln + l - 1] : 0.f;
    const float en0 =                en[b*Ln + l];
    const float enp = (l < Ln - 1) ? en[b*Ln + l + 1] : 0.f;
    const float pim = (l > 0)      ? pi[b*Ln + l - 1] : 0.f;
    const float pi0 =                pi[b*Ln + l];
    const float pip = (l < Ln - 1) ? pi[b*Ln + l + 1] : 0.f;

    float partial = 0.f;
    for (int d = threadIdx.x; d < Dn; d += 256) {
        const float de = wd[d*3+0]*dfm + wd[d*3+1]*df0 + wd[d*3+2]*dfp + bd[d];
        const float ee = we[d*3+0]*enm + we[d*3+1]*en0 + we[d*3+2]*enp + be[d];
        const float pe = wp[d*3+0]*pim + wp[d*3+1]*pi0 + wp[d*3+2]*pip + bp[d];
        const float xv = x[(size_t)bl * Dn + d] + ee + pe;
        xnew[(size_t)bl * Dn + d] = xv;
        partial += (xv + de) * wlin[d];
    }
    red[threadIdx.x] = partial;
    __syncthreads();
    for (int s = 128; s > 0; s >>= 1) {
        if ((int)threadIdx.x < s) red[threadIdx.x] += red[threadIdx.x + s];
        __syncthreads();
    }
    if (threadIdx.x == 0) {
        const float sv = red[0] + blin[0];
        // numerically stable softplus
        const float sp = fmaxf(sv, 0.f) + log1pf(expf(-fabsf(sv)));
        const float r  = (l < lens[b]) ? sp : 1.0f;
        inv_r[bl] = 1.f / r;
        log_r[bl] = logf(r);
    }
}

// ---------------------------------------------------------------------------
// K2: means[b,l] = 0.5*d + exclusive_cumsum(durations_int)  (tiny)
// ---------------------------------------------------------------------------
__global__ void k2_means(const int* __restrict__ di, float* __restrict__ means)
{
    const int b = threadIdx.x;
    if (b < Bn) {
        int cum = 0;
        for (int l = 0; l < Ln; ++l) {
            const int d = di[b*Ln + l];
            means[b*Ln + l] = 0.5f * (float)d + (float)cum;
            cum += d;
        }
    }
}

// ---------------------------------------------------------------------------
// K3: inv_colsum[b,t] = 1 / (sum_l probs[b,l,t] + 1e-20)   (no big stores)
// ---------------------------------------------------------------------------
__global__ __launch_bounds__(256)
void k3_colsum(const float* __restrict__ means,
               const float* __restrict__ inv_r,
               const float* __restrict__ log_r,
               const int*   __restrict__ lens,
               float* __restrict__ inv_cs, int T)
{
    const int b = blockIdx.y;
    const int t = blockIdx.x * 256 + threadIdx.x;
    if (t >= T) return;
    const int   len = lens[b];
    const float ft  = (float)t + 0.5f;
    float sum = 0.f;
    for (int l = 0; l < len; ++l) {           // invalid l contribute 0
        const float z = (ft - means[b*Ln + l]) * inv_r[b*Ln + l];
        sum += expf(-0.5f * z * z - log_r[b*Ln + l] - 0.5f * LOG2PI);
    }
    inv_cs[(size_t)b * T + t] = 1.f / (sum + 1e-20f);
}

// ---------------------------------------------------------------------------
// K3b: weights[b,l,t] = probs * inv_colsum  -> written directly to d_out
// ---------------------------------------------------------------------------
__global__ __launch_bounds__(256)
void k3b_weights(const float* __restrict__ means,
                 const float* __restrict__ inv_r,
                 const float* __restrict__ log_r,
                 const int*   __restrict__ lens,
                 const float* __restrict__ inv_cs,
                 float* __restrict__ wout, int T)
{
    const int b = blockIdx.z;
    const int l = blockIdx.y;
    const int t = blockIdx.x * 256 + threadIdx.x;
    if (t >= T) return;
    float w = 0.f;
    if (l < lens[b]) {
        const float ft = (float)t + 0.5f;
        const float z  = (ft - means[b*Ln + l]) * inv_r[b*Ln + l];
        w = expf(-0.5f * z * z - log_r[b*Ln + l] - 0.5f * LOG2PI)
            * inv_cs[(size_t)b * T + t];
    }
    wout[((size_t)(b * Ln + l)) * T + t] = w;
}

// ---------------------------------------------------------------------------
// K4: out[b,t,d] = sum_l W[b,l,t] * X[b,l,d]  via V_WMMA_F32_16X16X4_F32
// A(m=t, k=l) = W[l*T + t] (column-major T x L, ld = T)
// B(k=l, n=d) = X[l*Dn + d]
// Block = one 16(t) x 512(d) stripe: t-tile from blockIdx.x (uniform -> scalar
// epilogue branch), 8 waves cover the 8 64-wide d-tiles and share the A rows.
// Main K loop has unconditional W prefetch (global_prefetch_b8); short tail
// loop drops it, so both loops unroll cleanly with no in-loop branch.
// Non-temporal stores keep the streamed output out of L2 so W/X stay hot.
// ---------------------------------------------------------------------------
__global__ __launch_bounds__(256)
void k4_wmma_gemm(const float* __restrict__ W,   // (B, L, T) normalized weights
                  const float* __restrict__ X,   // (B, L, D) x_new
                  float* __restrict__ out,       // (B, T, D)
                  int T)
{
    const int lane = threadIdx.x & 31;
    const int wave = threadIdx.x >> 5;           // 0..7 -> d-tile
    const int b    = blockIdx.y;
    const int t0   = blockIdx.x * 16;            // block-uniform t-tile
    const int d0   = wave * 64;

    const int half   = lane >> 4;                // 0: K={k,k+1}, 1: K={k+2,k+3}
    const int lane16 = lane & 15;
    const int m      = t0 + lane16;              // t-row of A/C for this lane
    const int mA     = (m < T) ? m : 0;          // clamp loads; bad rows discarded

    const float* Wb = W + (size_t)b * Ln * T;
    const float* Xb = X + (size_t)b * Ln * Dn;

    v8f c0 = {}; v8f c1 = {}; v8f c2 = {}; v8f c3 = {};

    constexpr int PF = 16;                       // k-prefetch distance

    #define K4_BODY(kk)                                                        \
        do {                                                                   \
            const int ka = (kk) + 2 * half;                                    \
            v2f a;                                                             \
            a.x = Wb[(size_t)ka       * T + mA];                               \
            a.y = Wb[(size_t)(ka + 1) * T + mA];                               \
            const float* xr0 = Xb + (size_t)ka       * Dn + d0 + lane16;       \
            const float* xr1 = Xb + (size_t)(ka + 1) * Dn + d0 + lane16;       \
            v2f b0; b0.x = xr0[0];  b0.y = xr1[0];                             \
            v2f b1; b1.x = xr0[16]; b1.y = xr1[16];                            \
            v2f b2; b2.x = xr0[32]; b2.y = xr1[32];                            \
            v2f b3; b3.x = xr0[48]; b3.y = xr1[48];                            \
            c0 = __builtin_amdgcn_wmma_f32_16x16x4_f32(false, a, false, b0,    \
                                                       (short)0, c0, false, false); \
            c1 = __builtin_amdgcn_wmma_f32_16x16x4_f32(false, a, false, b1,    \
                                                       (short)0, c1, false, false); \
            c2 = __builtin_amdgcn_wmma_f32_16x16x4_f32(false, a, false, b2,    \
                                                       (short)0, c2, false, false); \
            c3 = __builtin_amdgcn_wmma_f32_16x16x4_f32(false, a, false, b3,    \
                                                       (short)0, c3, false, false); \
        } while (0)

    // main loop: unconditional prefetch of the cold W (A) stream
    #pragma unroll 2
    for (int k = 0; k < Ln - PF; k += 4) {
        const int kp = k + PF + 2 * half;
        __builtin_prefetch(&Wb[(size_t)kp       * T + mA], 0, 1);
        __builtin_prefetch(&Wb[(size_t)(kp + 1) * T + mA], 0, 1);
        K4_BODY(k);
    }
    // tail loop: no prefetch, static 4 iterations
    #pragma unroll
    for (int k = Ln - PF; k < Ln; k += 4) {
        K4_BODY(k);
    }
    #undef K4_BODY

    // C/D layout: VGPR v -> row (v + 8*half), col = lane16
    float* outb = out + (size_t)b * T * Dn;
    if (t0 + 16 <= T) {                          // scalar branch (blockIdx-based)
        #pragma unroll
        for (int v = 0; v < 8; ++v) {
            const int trow = t0 + v + 8 * half;
            float* o = outb + (size_t)trow * Dn + d0 + lane16;
            __builtin_nontemporal_store(c0[v], o);
            __builtin_nontemporal_store(c1[v], o + 16);
            __builtin_nontemporal_store(c2[v], o + 32);
            __builtin_nontemporal_store(c3[v], o + 48);
        }
    } else {
        // tail t-tile: per-row guard
        #pragma unroll
        for (int v = 0; v < 8; ++v) {
            const int trow = t0 + v + 8 * half;
            if (trow < T) {
                float* o = outb + (size_t)trow * Dn + d0 + lane16;
                o[0]  = c0[v];
                o[16] = c1[v];
                o[32] = c2[v];
                o[48] = c3[v];
            }
        }
    }
}

// ---------------------------------------------------------------------------
extern "C" void kernel_launch(void* const* d_in, const int* in_sizes, int n_in,
                              void* d_out, int out_size, void* d_ws, size_t ws_size,
                              hipStream_t stream)
{
    const float* x    = (const float*)d_in[0];
    const float* df   = (const float*)d_in[1];
    const int*   di   = (const int*)  d_in[2];
    const float* en   = (const float*)d_in[3];
    const float* pi   = (const float*)d_in[4];
    const int*   lens = (const int*)  d_in[5];
    const float* wd   = (const float*)d_in[6];
    const float* bd   = (const float*)d_in[7];
    const float* we   = (const float*)d_in[8];
    const float* be   = (const float*)d_in[9];
    const float* wp   = (const float*)d_in[10];
    const float* bp   = (const float*)d_in[11];
    const float* wlin = (const float*)d_in[12];
    const float* blin = (const float*)d_in[13];

    // out = x_upsamp (B*T*D) ++ weights (B*L*T)  =>  out_size = T * B * (D + L)
    const int T = out_size / (Bn * (Dn + Ln));

    float* out_x = (float*)d_out;                        // (B, T, D)
    float* out_w = (float*)d_out + (size_t)Bn * T * Dn;  // (B, L, T)

    // workspace layout (floats)
    float* ws    = (float*)d_ws;
    float* xnew  = ws;  ws += (size_t)Bn * Ln * Dn;      // 8M floats
    float* inv_r = ws;  ws += Bn * Ln;
    float* log_r = ws;  ws += Bn * Ln;
    float* means = ws;  ws += Bn * Ln;
    float* invcs = ws;                                   // Bn * T

    k1_features<<<Bn * Ln, 256, 0, stream>>>(x, df, en, pi, lens,
                                             wd, bd, we, be, wp, bp, wlin, blin,
                                             xnew, inv_r, log_r);

    k2_means<<<1, 32, 0, stream>>>(di, means);

    dim3 g3((T + 255) / 256, Bn);
    k3_colsum<<<g3, 256, 0, stream>>>(means, inv_r, log_r, lens, invcs, T);

    dim3 g3b((T + 255) / 256, Ln, Bn);
    k3b_weights<<<g3b, 256, 0, stream>>>(means, inv_r, log_r, lens, invcs, out_w, T);

    const int ntt = (T + 15) / 16;               // one block per t-tile
    dim3 g4(ntt, Bn);
    k4_wmma_gemm<<<g4, 256, 0, stream>>>(out_w, xnew, out_x, T);
}